// ClusterFPS_58437325029829
// MI455X (gfx1250) — compile-verified
//
#include <hip/hip_runtime.h>
#include <stdint.h>

#define NBATCH   16
#define NPTS     65536
#define NCENT    16
#define KM_ITERS 10
#define MAXNBR   256
#define NPG      128
#define BIGF     1.0e10f

typedef __attribute__((ext_vector_type(2))) float v2f;
typedef __attribute__((ext_vector_type(8))) float v8f;

// ---- workspace layout (byte offsets into d_ws) ----
#define WS_CENTERS 0          // NBATCH*NCENT*4 floats: x,y,z,|c|^2       (4 KB)
#define WS_ACCUM   4096       // NBATCH*NCENT*4 floats: sx,sy,sz,cnt      (4 KB)
#define WS_FAR     8192       // NBATCH*NCENT uint: initial FPS indices   (1 KB)
#define WS_NBR     12288      // NBATCH*NCENT*MAXNBR*3 floats             (768 KB)

// ---------------- JAX threefry2x32 (exact) ----------------
__device__ __forceinline__ unsigned rotl32(unsigned v, int d) {
  return (v << d) | (v >> (32 - d));
}
__device__ void tf2x32(unsigned k0, unsigned k1, unsigned x0, unsigned x1,
                       unsigned &o0, unsigned &o1) {
  const unsigned ks2 = k0 ^ k1 ^ 0x1BD11BDAu;
  const int ra[4] = {13, 15, 26, 6};
  const int rb[4] = {17, 29, 16, 24};
  x0 += k0; x1 += k1;
  #pragma unroll
  for (int i = 0; i < 4; ++i) { x0 += x1; x1 = rotl32(x1, ra[i]); x1 ^= x0; }
  x0 += k1; x1 += ks2 + 1u;
  #pragma unroll
  for (int i = 0; i < 4; ++i) { x0 += x1; x1 = rotl32(x1, rb[i]); x1 ^= x0; }
  x0 += ks2; x1 += k0 + 2u;
  #pragma unroll
  for (int i = 0; i < 4; ++i) { x0 += x1; x1 = rotl32(x1, ra[i]); x1 ^= x0; }
  x0 += k0; x1 += k1 + 3u;
  #pragma unroll
  for (int i = 0; i < 4; ++i) { x0 += x1; x1 = rotl32(x1, rb[i]); x1 ^= x0; }
  x0 += k1; x1 += ks2 + 4u;
  #pragma unroll
  for (int i = 0; i < 4; ++i) { x0 += x1; x1 = rotl32(x1, ra[i]); x1 ^= x0; }
  x0 += ks2; x1 += k0 + 5u;
  o0 = x0; o1 = x1;
}

// ---------------- kernel 1: RNG + init centers + zero accum ----------------
__global__ void k_init(const float* __restrict__ x, unsigned char* ws) {
  float*    centers = (float*)(ws + WS_CENTERS);
  float*    accum   = (float*)(ws + WS_ACCUM);
  unsigned* farr    = (unsigned*)(ws + WS_FAR);
  __shared__ unsigned s_kkm[2], s_kfps[2], s_idx[16];
  const int tid = threadIdx.x;

  if (tid == 0) {
    // key(42) = (0,42); split -> counters [0,1,2,3], cipher pairs (0,2),(1,3)
    unsigned a0, a1, b0, b1;
    tf2x32(0u, 42u, 0u, 2u, a0, b0);
    tf2x32(0u, 42u, 1u, 3u, a1, b1);
    s_kkm[0] = a0; s_kkm[1] = a1;   // k_km
    s_kfps[0] = b0; s_kfps[1] = b1; // k_fps
  }
  __syncthreads();
  if (tid < 8) {
    // randint(k_km,(16,),0,65536): span=2^16 -> bits & 0xFFFF
    unsigned o0, o1;
    tf2x32(s_kkm[0], s_kkm[1], (unsigned)tid, (unsigned)(tid + 8), o0, o1);
    s_idx[tid] = o0 & 0xFFFFu;
    s_idx[tid + 8] = o1 & 0xFFFFu;
  }
  if (tid < 128) {
    // randint(k_fps,(16,16),0,256): span=2^8 -> bits & 0xFF
    unsigned o0, o1;
    tf2x32(s_kfps[0], s_kfps[1], (unsigned)tid, (unsigned)(tid + 128), o0, o1);
    farr[tid] = o0 & 0xFFu;
    farr[tid + 128] = o1 & 0xFFu;
  }
  __syncthreads();
  // gather init centers: centers[b][j] = x[b][init_idx[j]]  (idx shared over b)
  {
    const int b = tid >> 4, j = tid & 15;
    const unsigned p = s_idx[j];
    const float* xp = x + ((size_t)b * NPTS + p) * 3;
    const float cx = xp[0], cy = xp[1], cz = xp[2];
    float* c = centers + (b * NCENT + j) * 4;
    c[0] = cx; c[1] = cy; c[2] = cz; c[3] = cx * cx + cy * cy + cz * cz;
    float* a = accum + (b * NCENT + j) * 4;
    a[0] = 0.f; a[1] = 0.f; a[2] = 0.f; a[3] = 0.f;
  }
}

// ---------------- kernel 2: WMMA distance + argmin + segment reduce ----------------
// grid = NBATCH*16 blocks, 256 threads (8 waves). Each wave: 32 tiles of 16 points.
// D = centers(16x4) x points(4x16): M=center, N=point -> per-point scores land in
// one lane's 8 accumulators; argmin is in-register + a single xor-16 exchange.
__global__ void k_assign(const float* __restrict__ x, unsigned char* ws) {
  const float* centers = (const float*)(ws + WS_CENTERS);
  float*       accum   = (float*)(ws + WS_ACCUM);
  __shared__ float bsum[8][NCENT * 4];   // per-wave accumulator slices

  const int tid   = threadIdx.x;
  const int batch = blockIdx.x >> 4;
  const int chunk = blockIdx.x & 15;
  const int lane  = tid & 31;
  const int wave  = tid >> 5;
  const int j     = lane & 15;          // center row (A) / point column (B)
  const int lowhalf = (lane < 16);

  bsum[wave][lane] = 0.0f;
  bsum[wave][lane + 32] = 0.0f;
  __syncthreads();

  // A matrix: centers 16x4. lane<16 -> (cx,cy) at K0,K1 ; lane>=16 -> (cz,0)
  const float* cc = centers + (batch * NCENT + j) * 4;
  v2f am;
  am[0] = lowhalf ? cc[0] : cc[2];
  am[1] = lowhalf ? cc[1] : 0.0f;

  // per-lane |c|^2 for the 8 center rows this lane's accumulators cover
  const int mbase = lowhalf ? 0 : 8;
  float csqr[8];
  #pragma unroll
  for (int r = 0; r < 8; ++r)
    csqr[r] = centers[(batch * NCENT + mbase + r) * 4 + 3];

  const float* xb = x + (size_t)batch * NPTS * 3;
  const int tile0 = chunk * 256 + wave * 32;   // 4096 tiles per batch

  for (int t = 0; t < 32; ++t) {
    const int base = (tile0 + t) * 16;
    const float* xp = xb + (size_t)(base + j) * 3;
    __builtin_prefetch(xp + 16 * 3, 0, 3);     // next tile, keep near
    const float px = xp[0], py = xp[1], pz = xp[2];

    // B matrix: points 4x16. lane<16 -> (x,y) rows K0,K1 ; lane>=16 -> (z,0)
    v2f bm;
    bm[0] = lowhalf ? px : pz;
    bm[1] = lowhalf ? py : 0.0f;

    v8f c = {0.f, 0.f, 0.f, 0.f, 0.f, 0.f, 0.f, 0.f};
    c = __builtin_amdgcn_wmma_f32_16x16x4_f32(false, am, false, bm,
                                              (short)0, c, false, false);

    // in-register argmin over this lane's 8 centers (strict < keeps lowest m)
    float sBest = fmaf(-2.0f, c[0], csqr[0]);
    int   mBest = mbase;
    #pragma unroll
    for (int r = 1; r < 8; ++r) {
      const float s = fmaf(-2.0f, c[r], csqr[r]);
      const int take = (int)(s < sBest);
      mBest = take ? (mbase + r) : mBest;
      sBest = take ? s : sBest;
    }
    // merge the two center halves for the same point (lane L <-> L^16)
    const float so = __shfl_xor(sBest, 16, 32);
    const int   mo = __shfl_xor(mBest, 16, 32);
    const int take = (int)(so < sBest) | ((int)(so == sBest) & (int)(mo < mBest));
    sBest = take ? so : sBest;
    mBest = take ? mo : mBest;

    // lanes 0..15 own point base+j's coords -> accumulate
    if (lowhalf) {
      float* bs = &bsum[wave][mBest * 4];
      atomicAdd(&bs[0], px);
      atomicAdd(&bs[1], py);
      atomicAdd(&bs[2], pz);
      atomicAdd(&bs[3], 1.0f);
    }
  }
  __syncthreads();
  if (tid < NCENT * 4) {
    float v = 0.0f;
    #pragma unroll
    for (int w = 0; w < 8; ++w) v += bsum[w][tid];
    atomicAdd(&accum[batch * NCENT * 4 + tid], v);
  }
}

// ---------------- kernel 3: center update ----------------
__global__ void k_update(unsigned char* ws) {
  float* centers = (float*)(ws + WS_CENTERS);
  float* accum   = (float*)(ws + WS_ACCUM);
  const int tid = threadIdx.x;           // 256 = NBATCH*NCENT
  float* a = accum + tid * 4;
  const float cnt = a[3];
  const float d = (cnt > 0.5f) ? cnt : 1.0f;
  const float cx = a[0] / d, cy = a[1] / d, cz = a[2] / d;
  float* c = centers + tid * 4;
  c[0] = cx; c[1] = cy; c[2] = cz; c[3] = cx * cx + cy * cy + cz * cz;
  a[0] = 0.f; a[1] = 0.f; a[2] = 0.f; a[3] = 0.f;
}

// ---------------- kernel 4: per-center top-256 (radix select in 256KB LDS) ----------------
// grid = NBATCH*NCENT blocks, 512 threads, dynamic LDS = NPTS*4 bytes.
__global__ void k_topk(const float* __restrict__ x, unsigned char* ws) {
  extern __shared__ float dlds[];                  // NPTS distances
  __shared__ unsigned hist[256];
  __shared__ unsigned long long sel[256];
  __shared__ unsigned sKth, sPrefix, sMask, sCntLess, sC0, sC1;

  const float* centers = (const float*)(ws + WS_CENTERS);
  float*       nbr     = (float*)(ws + WS_NBR);
  const int tid   = threadIdx.x;
  const int batch = blockIdx.x >> 4;
  const int k     = blockIdx.x & 15;

  const float* cc = centers + (batch * NCENT + k) * 4;
  const float cx = cc[0], cy = cc[1], cz = cc[2];
  const float* xb = x + (size_t)batch * NPTS * 3;

  for (int p = tid; p < NPTS; p += 512) {
    const float* xp = xb + (size_t)p * 3;
    const float dx = xp[0] - cx, dy = xp[1] - cy, dz = xp[2] - cz;
    dlds[p] = dx * dx + dy * dy + dz * dz;         // >=0 -> float bits monotone
  }
  if (tid == 0) { sKth = MAXNBR; sPrefix = 0; sMask = 0; sCntLess = 0; sC0 = 0; sC1 = 0; }
  __syncthreads();

  // 4x 8-bit radix passes (MSB first) to find the 256th smallest bit pattern
  for (int pass = 0; pass < 4; ++pass) {
    const int shift = 24 - 8 * pass;
    if (tid < 256) hist[tid] = 0;
    __syncthreads();
    const unsigned pref = sPrefix, msk = sMask;
    for (int p = tid; p < NPTS; p += 512) {
      const unsigned u = __float_as_uint(dlds[p]);
      if ((u & msk) == pref) atomicAdd(&hist[(u >> shift) & 255], 1u);
    }
    __syncthreads();
    if (tid == 0) {
      unsigned kth = sKth, cum = 0; int bin = 0;
      for (; bin < 256; ++bin) {
        const unsigned h = hist[bin];
        if (cum + h >= kth) break;
        cum += h;
      }
      sKth    = kth - cum;
      sPrefix = pref | ((unsigned)bin << shift);
      sMask   = msk | (0xFFu << shift);
    }
    __syncthreads();
  }
  const unsigned T = sPrefix;                      // exact 256th-smallest bits

  for (int p = tid; p < NPTS; p += 512)
    if (__float_as_uint(dlds[p]) < T) atomicAdd(&sCntLess, 1u);
  __syncthreads();
  const unsigned cntLess = sCntLess;
  const unsigned needEq  = MAXNBR - cntLess;

  for (int p = tid; p < NPTS; p += 512) {
    const unsigned u = __float_as_uint(dlds[p]);
    const unsigned long long key = ((unsigned long long)u << 32) | (unsigned)p;
    if (u < T) {
      const unsigned pos = atomicAdd(&sC0, 1u);
      sel[pos] = key;
    } else if (u == T) {
      const unsigned pos = atomicAdd(&sC1, 1u);
      if (pos < needEq) sel[cntLess + pos] = key;
    }
  }
  __syncthreads();

  // bitonic sort 256 (dist,idx) keys ascending -> stable argsort order
  for (int kk = 2; kk <= 256; kk <<= 1) {
    for (int jj = kk >> 1; jj > 0; jj >>= 1) {
      if (tid < 256) {
        const int ixj = tid ^ jj;
        if (ixj > tid) {
          const unsigned long long a = sel[tid], b = sel[ixj];
          const bool sw = (a > b) == ((tid & kk) == 0);
          sel[tid] = sw ? b : a;
          sel[ixj] = sw ? a : b;
        }
      }
      __syncthreads();
    }
  }

  if (tid < 256) {
    const unsigned p = (unsigned)(sel[tid] & 0xFFFFFFFFu);
    const float* xp = xb + (size_t)p * 3;
    float* o = nbr + ((size_t)(batch * NCENT + k) * MAXNBR + tid) * 3;
    o[0] = xp[0]; o[1] = xp[1]; o[2] = xp[2];
  }
}

// ---------------- kernel 5: FPS (256 pts -> 128 picks) + final gather ----------------
__global__ void k_fps(const unsigned char* ws, float* __restrict__ out) {
  const float*    nbr  = (const float*)(ws + WS_NBR);
  const unsigned* farr = (const unsigned*)(ws + WS_FAR);
  __shared__ float rv[256];
  __shared__ int   ri[256];
  __shared__ float scx, scy, scz;
  __shared__ int   sFar;
  __shared__ unsigned fidx[NPG];

  const int tid = threadIdx.x;
  const int bg  = blockIdx.x;                       // b*16+g
  const float* base = nbr + (size_t)bg * MAXNBR * 3;
  const float px = base[tid * 3 + 0];
  const float py = base[tid * 3 + 1];
  const float pz = base[tid * 3 + 2];
  float dist = BIGF;

  if (tid == 0) sFar = (int)farr[bg];
  __syncthreads();

  for (int it = 0; it < NPG; ++it) {
    const int far = sFar;
    if (tid == 0) fidx[it] = (unsigned)far;
    if (tid == far) { scx = px; scy = py; scz = pz; }
    __syncthreads();
    const float dx = px - scx, dy = py - scy, dz = pz - scz;
    dist = fminf(dist, dx * dx + dy * dy + dz * dz);
    rv[tid] = dist; ri[tid] = tid;
    __syncthreads();
    for (int s = 128; s > 0; s >>= 1) {
      if (tid < s) {
        const float v1 = rv[tid],     v2 = rv[tid + s];
        const int   i1 = ri[tid],     i2 = ri[tid + s];
        const int take = (int)(v2 > v1) | ((int)(v2 == v1) & (int)(i2 < i1));
        rv[tid] = take ? v2 : v1;
        ri[tid] = take ? i2 : i1;
      }
      __syncthreads();
    }
    if (tid == 0) sFar = ri[0];
    __syncthreads();
  }

  if (tid < NPG) {
    const unsigned gi = fidx[tid];
    const float* src = base + (size_t)gi * 3;
    float* o = out + ((size_t)bg * NPG + tid) * 3;
    o[0] = src[0]; o[1] = src[1]; o[2] = src[2];
  }
}

// ---------------- launcher ----------------
extern "C" void kernel_launch(void* const* d_in, const int* in_sizes, int n_in,
                              void* d_out, int out_size, void* d_ws, size_t ws_size,
                              hipStream_t stream) {
  (void)in_sizes; (void)n_in; (void)out_size; (void)ws_size;
  const float* x  = (const float*)d_in[0];
  unsigned char* ws = (unsigned char*)d_ws;
  float* out = (float*)d_out;

  // allow 256KB dynamic LDS for the top-k kernel (CDNA5: 320KB/WGP)
  (void)hipFuncSetAttribute((const void*)k_topk,
                            hipFuncAttributeMaxDynamicSharedMemorySize,
                            NPTS * (int)sizeof(float));

  k_init<<<1, 256, 0, stream>>>(x, ws);
  for (int it = 0; it < KM_ITERS; ++it) {
    k_assign<<<NBATCH * 16, 256, 0, stream>>>(x, ws);
    k_update<<<1, 256, 0, stream>>>(ws);
  }
  k_topk<<<NBATCH * NCENT, 512, NPTS * sizeof(float), stream>>>(x, ws);
  k_fps<<<NBATCH * NCENT, 256, 0, stream>>>(ws, out);
}